// TopologicalLoss_41077067218973
// MI455X (gfx1250) — compile-verified
//
#include <hip/hip_runtime.h>

typedef __attribute__((ext_vector_type(2))) float v2f;
typedef __attribute__((ext_vector_type(4))) float v4f;
typedef __attribute__((ext_vector_type(8))) float v8f;

#define DICE_EPS 1e-6f

// Wave-level 32-lane sum using V_WMMA_F32_16X16X4_F32.
// A (16x4): lane L<16 -> A[L][0]=s, A[L][1]=0 ; lane L>=16 -> A[L-16][2]=s, A[L-16][3]=0.
// B = ones (4x16). D[m][n] = s(m) + s(m+16), independent of n.
// D layout: lanes 0-15 hold rows 0-7 in d[0..7]; lanes 16-31 hold rows 8-15.
// => sum(d[0..7]) in any lane of 0-15 is Sum_{m=0..7}(s(m)+s(m+16)),
//    in any lane of 16-31 is Sum_{m=8..15}(s(m)+s(m+16)).
// Half-wave partials from lane 0 and lane 16 add to the full 32-lane sum.
// Requires EXEC all ones (call sites are fully converged, blockDim multiple of 32).
__device__ __forceinline__ float wave_halfsum_wmma(float s) {
    v2f a; a[0] = s;     a[1] = 0.0f;
    v2f b; b[0] = 1.0f;  b[1] = 1.0f;
    v8f c = {};
    v8f d = __builtin_amdgcn_wmma_f32_16x16x4_f32(
        /*neg_a=*/false, a, /*neg_b=*/false, b,
        /*c_mod=*/(short)0, c, /*reuse_a=*/false, /*reuse_b=*/false);
    float t = d[0];
    t += d[1]; t += d[2]; t += d[3];
    t += d[4]; t += d[5]; t += d[6]; t += d[7];
    return t;  // lanes 0-15: rows 0-7 total; lanes 16-31: rows 8-15 total
}

// Pass 1: per (block, channel) partial sums of x*t, x*x, t*t.
// grid = (G, 3), block = 256 (8 waves). NT b128 loads: data is single-use.
__global__ __launch_bounds__(256) void dice_partial_kernel(
    const float* __restrict__ x, const float* __restrict__ t,
    float* __restrict__ ws, long long perChannel, long long n4, int G) {
    const int c = blockIdx.y;
    const float* xc = x + (long long)c * perChannel;
    const float* tc = t + (long long)c * perChannel;
    const v4f* X = (const v4f*)xc;
    const v4f* T = (const v4f*)tc;

    long long gtid   = (long long)blockIdx.x * blockDim.x + threadIdx.x;
    long long stride = (long long)gridDim.x * blockDim.x;

    float sn = 0.0f, sxx = 0.0f, stt = 0.0f;
    for (long long i = gtid; i < n4; i += stride) {
        v4f xv = __builtin_nontemporal_load(&X[i]);
        v4f tv = __builtin_nontemporal_load(&T[i]);
        sn  = fmaf(xv[0], tv[0], sn);
        sn  = fmaf(xv[1], tv[1], sn);
        sn  = fmaf(xv[2], tv[2], sn);
        sn  = fmaf(xv[3], tv[3], sn);
        sxx = fmaf(xv[0], xv[0], sxx);
        sxx = fmaf(xv[1], xv[1], sxx);
        sxx = fmaf(xv[2], xv[2], sxx);
        sxx = fmaf(xv[3], xv[3], sxx);
        stt = fmaf(tv[0], tv[0], stt);
        stt = fmaf(tv[1], tv[1], stt);
        stt = fmaf(tv[2], tv[2], stt);
        stt = fmaf(tv[3], tv[3], stt);
    }
    // scalar tail (perChannel % 4 != 0 safety; divides exactly for this shape)
    for (long long e = n4 * 4 + gtid; e < perChannel; e += stride) {
        float xv = xc[e], tv = tc[e];
        sn  = fmaf(xv, tv, sn);
        sxx = fmaf(xv, xv, sxx);
        stt = fmaf(tv, tv, stt);
    }

    // Wave reduction on the matrix pipe (EXEC all ones here: no divergence).
    float hn  = wave_halfsum_wmma(sn);
    float hxx = wave_halfsum_wmma(sxx);
    float htt = wave_halfsum_wmma(stt);

    __shared__ float red[3][16];  // 8 waves x 2 half-wave slots
    unsigned lane = threadIdx.x & 31u;
    unsigned w    = threadIdx.x >> 5;
    if ((lane & 15u) == 0u) {
        unsigned slot = w * 2u + (lane >> 4);
        red[0][slot] = hn;
        red[1][slot] = hxx;
        red[2][slot] = htt;
    }
    __syncthreads();
    if (threadIdx.x == 0) {
        float a0 = 0.0f, a1 = 0.0f, a2 = 0.0f;
        #pragma unroll
        for (int i = 0; i < 16; ++i) { a0 += red[0][i]; a1 += red[1][i]; a2 += red[2][i]; }
        long long b = (long long)c * G + blockIdx.x;
        ws[b * 3 + 0] = a0;
        ws[b * 3 + 1] = a1;
        ws[b * 3 + 2] = a2;
    }
}

// Pass 2: single block folds G partials per channel, emits the scalar loss.
__global__ __launch_bounds__(256) void dice_final_kernel(
    const float* __restrict__ ws, float* __restrict__ out, int G) {
    float acc[9];
    #pragma unroll
    for (int q = 0; q < 9; ++q) acc[q] = 0.0f;
    for (int c = 0; c < 3; ++c) {
        for (int i = threadIdx.x; i < G; i += blockDim.x) {
            const float* p = ws + ((long long)c * G + i) * 3;
            acc[c * 3 + 0] += p[0];
            acc[c * 3 + 1] += p[1];
            acc[c * 3 + 2] += p[2];
        }
    }
    __shared__ float buf[256];
    __shared__ float res[9];
    for (int q = 0; q < 9; ++q) {
        buf[threadIdx.x] = acc[q];
        __syncthreads();
        for (int off = 128; off > 0; off >>= 1) {
            if ((int)threadIdx.x < off) buf[threadIdx.x] += buf[threadIdx.x + off];
            __syncthreads();
        }
        if (threadIdx.x == 0) res[q] = buf[0];
        __syncthreads();
    }
    if (threadIdx.x == 0) {
        float loss = 0.0f;
        #pragma unroll
        for (int c = 0; c < 3; ++c) {
            float num = res[c * 3 + 0];
            float den = res[c * 3 + 1] + res[c * 3 + 2];
            loss += -2.0f * num / fmaxf(den, DICE_EPS);
        }
        out[0] = loss;
    }
}

extern "C" void kernel_launch(void* const* d_in, const int* in_sizes, int n_in,
                              void* d_out, int out_size, void* d_ws, size_t ws_size,
                              hipStream_t stream) {
    const float* x = (const float*)d_in[0];
    const float* t = (const float*)d_in[1];
    float* out = (float*)d_out;
    float* ws  = (float*)d_ws;

    const int C = 3;
    long long total      = (long long)in_sizes[0];   // 1*3*96*256*256
    long long perChannel = total / C;                // 6,291,456
    long long n4         = perChannel / 4;           // 1,572,864 float4 per channel

    int G = 512;                                     // blocks per channel
    while (G > 1 && (size_t)C * G * 3 * sizeof(float) > ws_size) G >>= 1;

    dim3 grid((unsigned)G, (unsigned)C);
    dice_partial_kernel<<<grid, 256, 0, stream>>>(x, t, ws, perChannel, n4, G);
    dice_final_kernel<<<1, 256, 0, stream>>>(ws, out, G);
}